// ExpSparseGraphAttentionLayer_83459804496008
// MI455X (gfx1250) — compile-verified
//
#include <hip/hip_runtime.h>
#include <hip/hip_bf16.h>
#include <math.h>

#define BB 2
#define NN 512
#define FIN 128
#define FOUT 128
#define HH 8
#define FH 16
#define TOPM 16
#define SLOPE 0.2f

typedef float v2f __attribute__((ext_vector_type(2)));
typedef float v8f __attribute__((ext_vector_type(8)));

// ---------------------------------------------------------------------------
// Stage 1: g_l = h @ w_l, g_r = h @ w_r  via V_WMMA_F32_16X16X4_F32.
// One wave (32 lanes) per 16x16 output tile. M = B*N = 1024, K = 128, N = 128.
// A-frag layout (ISA 7.12.2, 32-bit A 16x4): lane<16 holds A[M=lane][K=k,k+1],
// lane>=16 holds A[M=lane-16][K=k+2,k+3]. B-frag mirrored. D: VGPR r holds
// D[M = r + 8*(lane>=16)][N = lane%16].
// ---------------------------------------------------------------------------
__global__ __launch_bounds__(32) void gat_gemm_wmma(
    const float* __restrict__ hmat,   // (1024, 128)
    const float* __restrict__ wl,     // (128, 128)
    const float* __restrict__ wr,     // (128, 128)
    float* __restrict__ gl,           // (1024, 128)
    float* __restrict__ gr)           // (1024, 128)
{
    const int tm   = blockIdx.x;          // 0..63  (M tile)
    const int tn   = blockIdx.y;          // 0..7   (N tile)
    const int lane = threadIdx.x;         // 0..31
    const int half = lane >> 4;           // 0 / 1
    const int l16  = lane & 15;

    const int arow = tm * 16 + l16;       // A row for this lane
    const int bcol = tn * 16 + l16;       // B/D column for this lane

    v8f accL = {};
    v8f accR = {};

    #pragma unroll 8
    for (int k = 0; k < FIN; k += 4) {
        const int ka = k + 2 * half;      // this lane-half's K pair
        v2f a, bl, br;
        a.x  = hmat[arow * FIN + ka];
        a.y  = hmat[arow * FIN + ka + 1];
        bl.x = wl[(ka)     * FOUT + bcol];
        bl.y = wl[(ka + 1) * FOUT + bcol];
        br.x = wr[(ka)     * FOUT + bcol];
        br.y = wr[(ka + 1) * FOUT + bcol];
        accL = __builtin_amdgcn_wmma_f32_16x16x4_f32(
            false, a, false, bl, (short)0, accL, false, false);
        accR = __builtin_amdgcn_wmma_f32_16x16x4_f32(
            false, a, false, br, (short)0, accR, false, false);
    }

    #pragma unroll
    for (int r = 0; r < 8; ++r) {
        const int drow = tm * 16 + r + 8 * half;
        gl[drow * FOUT + bcol] = accL[r];
        gr[drow * FOUT + bcol] = accR[r];
    }
}

// ---------------------------------------------------------------------------
// Stage 2: fused scores + top-16 + softmax + aggregation + final lrelu.
// One workgroup per (b, h, i-block of 128). LDS stages the (512 x 16) slices
// of g_l and g_r for this (b,h): all lanes walk the same j -> LDS broadcast.
// Each thread owns one destination node i: keeps a sorted 16-entry top list
// in registers (branch-free unrolled insertion, strict '>' preserves top_k
// first-index tie-breaking), then softmax + weighted sum of g_r rows.
// ---------------------------------------------------------------------------
__global__ __launch_bounds__(128) void gat_score_topk_agg(
    const float* __restrict__ gl,       // (B,N,H,FH) = (1024,128)
    const float* __restrict__ gr,
    const int*   __restrict__ adj,      // (B,N,N)
    const float* __restrict__ attn_w,   // (FH)
    const float* __restrict__ attn_b,   // (1)
    float*       __restrict__ out)      // (B,N,H*FH)
{
    __shared__ float lgl[NN][FH];       // 32 KB
    __shared__ float lgr[NN][FH];       // 32 KB

    const int bid  = blockIdx.x;        // 0..63
    const int iblk = bid & 3;           // N/128 = 4 blocks
    const int hh   = (bid >> 2) & 7;
    const int b    = bid >> 5;
    const int tid  = threadIdx.x;

    // Stage g_l / g_r slices for this (b,h): 512 rows x 4 float4's each.
    for (int idx = tid; idx < NN * (FH / 4); idx += 128) {
        const int j  = idx >> 2;
        const int f4 = (idx & 3) << 2;
        const int gsrc = ((b * NN + j) * HH + hh) * FH + f4;
        *(float4*)&lgl[j][f4] = *(const float4*)&gl[gsrc];
        *(float4*)&lgr[j][f4] = *(const float4*)&gr[gsrc];
    }
    __syncthreads();

    float aw[FH];
    #pragma unroll
    for (int f = 0; f < FH; ++f) aw[f] = attn_w[f];
    const float ab = attn_b[0];

    const int i = iblk * 128 + tid;     // destination node
    float gri[FH];
    #pragma unroll
    for (int f = 0; f < FH; ++f) gri[f] = lgr[i][f];

    const int* __restrict__ adjrow = adj + ((size_t)b * NN + i) * NN;

    const float NEG_INF = -__builtin_inff();
    float tv[TOPM];
    int   ti[TOPM];
    #pragma unroll
    for (int k = 0; k < TOPM; ++k) { tv[k] = NEG_INF; ti[k] = 0; }

    for (int j = 0; j < NN; ++j) {
        const int a = adjrow[j];
        float s = ab;
        #pragma unroll
        for (int f = 0; f < FH; ++f) {
            float x = gri[f] + lgl[j][f];
            x = fmaxf(x, SLOPE * x);          // leaky-relu (slope < 1)
            s = __builtin_fmaf(x, aw[f], s);
        }
        s = (a == 0) ? NEG_INF : s;
        if (s > tv[TOPM - 1]) {               // insert into sorted top list
            float v = s; int id = j;
            #pragma unroll
            for (int k = 0; k < TOPM; ++k) {
                const bool gt = v > tv[k];
                const float nv = gt ? v : tv[k];
                const int   ni = gt ? id : ti[k];
                const float ov = gt ? tv[k] : v;
                const int   oi = gt ? ti[k] : id;
                tv[k] = nv; ti[k] = ni; v = ov; id = oi;
            }
        }
    }

    // softmax over the 16 kept entries (-inf entries -> weight 0)
    const float m = tv[0];
    float w[TOPM];
    float den = 0.f;
    #pragma unroll
    for (int k = 0; k < TOPM; ++k) { w[k] = __expf(tv[k] - m); den += w[k]; }
    const float inv = 1.0f / den;

    float acc[FH];
    #pragma unroll
    for (int f = 0; f < FH; ++f) acc[f] = 0.f;
    #pragma unroll
    for (int k = 0; k < TOPM; ++k) {
        const float wk = w[k] * inv;
        const int   j  = ti[k];
        #pragma unroll
        for (int f = 0; f < FH; ++f)
            acc[f] = __builtin_fmaf(wk, lgr[j][f], acc[f]);
    }

    float* __restrict__ dst = out + ((size_t)b * NN + i) * (HH * FH) + hh * FH;
    #pragma unroll
    for (int f = 0; f < FH; ++f) {
        float y = acc[f];
        y = fmaxf(y, SLOPE * y);              // final leaky-relu
        dst[f] = y;
    }
}

extern "C" void kernel_launch(void* const* d_in, const int* in_sizes, int n_in,
                              void* d_out, int out_size, void* d_ws, size_t ws_size,
                              hipStream_t stream) {
    const float* hmat   = (const float*)d_in[0];   // (B,N,F_IN) f32
    const int*   adj    = (const int*)  d_in[1];   // (B,N,N)    i32
    const float* wl     = (const float*)d_in[2];   // (F_IN,F_OUT)
    const float* wr     = (const float*)d_in[3];   // (F_IN,F_OUT)
    const float* attn_w = (const float*)d_in[4];   // (FH)
    const float* attn_b = (const float*)d_in[5];   // scalar

    float* gl = (float*)d_ws;                      // (1024,128)
    float* gr = gl + (size_t)BB * NN * FOUT;       // (1024,128)

    dim3 g1((BB * NN) / 16, FOUT / 16);            // 64 x 8 tiles, 1 wave each
    gat_gemm_wmma<<<g1, 32, 0, stream>>>(hmat, wl, wr, gl, gr);

    gat_score_topk_agg<<<BB * HH * (NN / 128), 128, 0, stream>>>(
        gl, gr, adj, attn_w, attn_b, (float*)d_out);
}